// PromptPool_83099027243778
// MI455X (gfx1250) — compile-verified
//
#include <hip/hip_runtime.h>
#include <math.h>

typedef float v2f __attribute__((ext_vector_type(2)));
typedef float v8f __attribute__((ext_vector_type(8)));

#define B_DIM   4096
#define P_DIM   512
#define D_DIM   1536
#define L_DIM   8
#define H_DIM   768
#define TOPK    4
#define ROW_FLOATS (L_DIM * H_DIM)   // 6144 floats = 24 KB per gathered prompt
#define NEG_INF (-__builtin_inff())

// ---------------------------------------------------------------------------
// Normalize each row of prompts_key into knorm (512 rows x 1536)
// ---------------------------------------------------------------------------
__global__ void __launch_bounds__(256)
knorm_kernel(const float* __restrict__ in, float* __restrict__ out) {
    __shared__ float red[256];
    const int row = blockIdx.x;
    const float* src = in + (size_t)row * D_DIM;
    float acc = 0.f;
    for (int i = threadIdx.x; i < D_DIM; i += 256) { float x = src[i]; acc += x * x; }
    red[threadIdx.x] = acc; __syncthreads();
    for (int s = 128; s > 0; s >>= 1) {
        if (threadIdx.x < s) red[threadIdx.x] += red[threadIdx.x + s];
        __syncthreads();
    }
    const float inv = 1.0f / fmaxf(sqrtf(red[0]), 1e-12f);
    float* dst = out + (size_t)row * D_DIM;
    for (int i = threadIdx.x; i < D_DIM; i += 256) dst[i] = src[i] * inv;
}

// ---------------------------------------------------------------------------
// Per-row inverse norm of querys (4096 rows x 1536) -> invq[4096]
// ---------------------------------------------------------------------------
__global__ void __launch_bounds__(256)
qinv_kernel(const float* __restrict__ in, float* __restrict__ invq) {
    __shared__ float red[256];
    const int row = blockIdx.x;
    const float* src = in + (size_t)row * D_DIM;
    float acc = 0.f;
    for (int i = threadIdx.x; i < D_DIM; i += 256) { float x = src[i]; acc += x * x; }
    red[threadIdx.x] = acc; __syncthreads();
    for (int s = 128; s > 0; s >>= 1) {
        if (threadIdx.x < s) red[threadIdx.x] += red[threadIdx.x + s];
        __syncthreads();
    }
    if (threadIdx.x == 0) invq[row] = 1.0f / fmaxf(sqrtf(red[0]), 1e-12f);
}

// ---------------------------------------------------------------------------
// sim[4096x512] = Q[4096x1536] * Knorm^T[1536x512] via V_WMMA_F32_16X16X4_F32.
// One wave32 per 16x64 output strip: A tile (16x4) reused across 4 N-tiles.
// A layout: lanes 0-15 -> M=lane, K={k,k+1}; lanes 16-31 -> M=lane-16, K={k+2,k+3}
// B layout mirrors with N in place of M (B column j = Knorm row j).
// ---------------------------------------------------------------------------
__global__ void __launch_bounds__(32)
gemm_wmma_kernel(const float* __restrict__ Q, const float* __restrict__ Kn,
                 float* __restrict__ sim) {
    const int lane = threadIdx.x;
    const bool hi  = lane >= 16;
    const int  lr  = lane & 15;
    const int  m0  = blockIdx.x * 16;
    const int  n0  = blockIdx.y * 64;
    const int  kof = hi ? 2 : 0;

    const float* ap  = Q  + (size_t)(m0 + lr) * D_DIM + kof;
    const float* bp0 = Kn + (size_t)(n0 +  0 + lr) * D_DIM + kof;
    const float* bp1 = Kn + (size_t)(n0 + 16 + lr) * D_DIM + kof;
    const float* bp2 = Kn + (size_t)(n0 + 32 + lr) * D_DIM + kof;
    const float* bp3 = Kn + (size_t)(n0 + 48 + lr) * D_DIM + kof;

    v8f c0 = {}, c1 = {}, c2 = {}, c3 = {};

#pragma unroll 4
    for (int k = 0; k < D_DIM; k += 4) {
        v2f a  = *(const v2f*)(ap  + k);
        v2f b0 = *(const v2f*)(bp0 + k);
        v2f b1 = *(const v2f*)(bp1 + k);
        v2f b2 = *(const v2f*)(bp2 + k);
        v2f b3 = *(const v2f*)(bp3 + k);
        c0 = __builtin_amdgcn_wmma_f32_16x16x4_f32(false, a, false, b0, (short)0, c0, false, false);
        c1 = __builtin_amdgcn_wmma_f32_16x16x4_f32(false, a, false, b1, (short)0, c1, false, false);
        c2 = __builtin_amdgcn_wmma_f32_16x16x4_f32(false, a, false, b2, (short)0, c2, false, false);
        c3 = __builtin_amdgcn_wmma_f32_16x16x4_f32(false, a, false, b3, (short)0, c3, false, false);
    }

    // C/D layout: VGPR r, lanes 0-15 -> M=r, lanes 16-31 -> M=r+8; N = lane&15
#pragma unroll
    for (int r = 0; r < 8; ++r) {
        const int row = m0 + r + (hi ? 8 : 0);
        float* dst = sim + (size_t)row * P_DIM + n0 + lr;
        dst[0]  = c0[r];
        dst[16] = c1[r];
        dst[32] = c2[r];
        dst[48] = c3[r];
    }
}

// ---------------------------------------------------------------------------
// Per-row top-4 (values + indices, jax tie-break: lower index first) + row sum.
// Scale by invq here (positive scale preserves ordering).
// ---------------------------------------------------------------------------
__global__ void __launch_bounds__(256)
topk_kernel(const float* __restrict__ sim, const float* __restrict__ invq,
            float* __restrict__ simtop, int* __restrict__ idxout,
            float* __restrict__ rowsum) {
    const int b = blockIdx.x * 256 + threadIdx.x;
    if (b >= B_DIM) return;
    const float s = invq[b];
    const float* row = sim + (size_t)b * P_DIM;
    float v0 = NEG_INF, v1 = NEG_INF, v2 = NEG_INF, v3 = NEG_INF;
    int   i0 = 0, i1 = 0, i2 = 0, i3 = 0;
    float sum = 0.f;
    for (int p = 0; p < P_DIM; ++p) {
        const float x = row[p] * s;
        sum += x;
        if (x > v3) {
            if (x > v0)      { v3=v2;i3=i2; v2=v1;i2=i1; v1=v0;i1=i0; v0=x;i0=p; }
            else if (x > v1) { v3=v2;i3=i2; v2=v1;i2=i1; v1=x;i1=p; }
            else if (x > v2) { v3=v2;i3=i2; v2=x;i2=p; }
            else             { v3=x;i3=p; }
        }
    }
    simtop[b*TOPK+0] = v0; simtop[b*TOPK+1] = v1;
    simtop[b*TOPK+2] = v2; simtop[b*TOPK+3] = v3;
    idxout[b*TOPK+0] = i0; idxout[b*TOPK+1] = i1;
    idxout[b*TOPK+2] = i2; idxout[b*TOPK+3] = i3;
    rowsum[b] = sum;
}

// ---------------------------------------------------------------------------
// Deterministic fixed-order reduction of 4096 row sums -> loss scalar
// ---------------------------------------------------------------------------
__global__ void __launch_bounds__(256)
loss_kernel(const float* __restrict__ rowsum, float* __restrict__ loss) {
    __shared__ float red[256];
    float acc = 0.f;
    for (int i = threadIdx.x; i < B_DIM; i += 256) acc += rowsum[i];
    red[threadIdx.x] = acc; __syncthreads();
    for (int s = 128; s > 0; s >>= 1) {
        if (threadIdx.x < s) red[threadIdx.x] += red[threadIdx.x + s];
        __syncthreads();
    }
    if (threadIdx.x == 0) loss[0] = -red[0] / ((float)B_DIM * (float)P_DIM);
}

// ---------------------------------------------------------------------------
// Gather: selected[b,j,:,:] = prompts[idx[b,j],:,:]  (24 KB per copy, b128 ops)
// One 256-thread block per (b,j) pair; 6 float4 per thread. HBM bound: ~810 MB.
// ---------------------------------------------------------------------------
__global__ void __launch_bounds__(256)
gather_kernel(const float* __restrict__ prompts, const int* __restrict__ idx,
              float* __restrict__ out) {
    const int pair = blockIdx.x;                 // 0 .. 16383
    const int p = idx[pair];
    const float4* src = (const float4*)(prompts + (size_t)p * ROW_FLOATS);
    float4* dst = (float4*)(out + (size_t)pair * ROW_FLOATS);
#pragma unroll
    for (int i = threadIdx.x; i < ROW_FLOATS / 4; i += 256) dst[i] = src[i];
}

// ---------------------------------------------------------------------------
extern "C" void kernel_launch(void* const* d_in, const int* in_sizes, int n_in,
                              void* d_out, int out_size, void* d_ws, size_t ws_size,
                              hipStream_t stream) {
    (void)in_sizes; (void)n_in; (void)out_size; (void)ws_size;

    const float* querys  = (const float*)d_in[0];   // [4096,1536]
    const float* pkey    = (const float*)d_in[1];   // [512,1536]
    const float* prompts = (const float*)d_in[2];   // [512,8,768]
    // d_in[3] = topk (=4), compile-time constant here

    float* out      = (float*)d_out;
    float* selected = out;                                             // 4096*4*8*768
    float* simtop   = out + (size_t)B_DIM * TOPK * ROW_FLOATS;         // 4096*4
    float* loss     = simtop + (size_t)B_DIM * TOPK;                   // 1

    float* knorm  = (float*)d_ws;                                      // 512*1536
    float* invq   = knorm + (size_t)P_DIM * D_DIM;                     // 4096
    float* sim    = invq + B_DIM;                                      // 4096*512
    float* rowsum = sim + (size_t)B_DIM * P_DIM;                       // 4096
    int*   idxbuf = (int*)(rowsum + B_DIM);                            // 4096*4

    knorm_kernel<<<P_DIM, 256, 0, stream>>>(pkey, knorm);
    qinv_kernel <<<B_DIM, 256, 0, stream>>>(querys, invq);

    dim3 ggrid(B_DIM / 16, P_DIM / 64);   // 256 x 8 waves, 16x64 strip each
    gemm_wmma_kernel<<<ggrid, 32, 0, stream>>>(querys, knorm, sim);

    topk_kernel<<<B_DIM / 256, 256, 0, stream>>>(sim, invq, simtop, idxbuf, rowsum);
    loss_kernel<<<1, 256, 0, stream>>>(rowsum, loss);
    gather_kernel<<<B_DIM * TOPK, 256, 0, stream>>>(prompts, idxbuf, selected);
}